// MultiscaleAttention_36919538876505
// MI455X (gfx1250) — compile-verified
//
#include <hip/hip_runtime.h>
#include <stdint.h>

#define LAYERS 12
#define DIM 768
#define TOKEN_ELEMS (LAYERS * DIM)        // 9216 floats = 36864 bytes per token
#define THREADS 256
#define WAVES (THREADS / 32)

#if defined(__has_builtin)
#if __has_builtin(__builtin_amdgcn_wmma_f32_16x16x4_f32)
#define USE_WMMA_MLP 1
#endif
#endif

typedef __attribute__((ext_vector_type(2))) float v2f;
typedef __attribute__((ext_vector_type(8))) float v8f;

// One token (b,s) per workgroup.
// Pipeline: async-stage X[token] (36KB) global->LDS, reduce over D per layer,
// 2-layer tanh MLP (12x12 via WMMA f32 16x16x4), softmax over layers,
// weighted combine from LDS.
__global__ __launch_bounds__(THREADS) void msattn_kernel(
    const float* __restrict__ X,
    const float* __restrict__ W1, const float* __restrict__ b1,
    const float* __restrict__ W2, const float* __restrict__ b2,
    float* __restrict__ out)
{
    __shared__ __align__(16) float tile[TOKEN_ELEMS];   // X[token] staged in LDS
    __shared__ float wpart[WAVES][LAYERS];              // per-wave partial sums
    __shared__ float xsum[LAYERS];
    __shared__ float h1s[LAYERS];
    __shared__ float wgt[LAYERS];

    const int tid  = threadIdx.x;
    const int lane = tid & 31;
    const int wave = tid >> 5;

    const size_t token = blockIdx.x;
    const float* __restrict__ src = X + token * (size_t)TOKEN_ELEMS;

    // ---- Stage token tile: 2304 x B128 async transfers (9 per thread) ----
    {
        uint32_t voff = (uint32_t)(tid * 16);                      // byte offset in tile
        uint32_t ldsa = (uint32_t)(uintptr_t)(&tile[0]) + voff;    // LDS byte address
        #pragma unroll
        for (int i = 0; i < 9; ++i) {
            asm volatile(
                "global_load_async_to_lds_b128 %0, %1, %2"
                :
                : "v"(ldsa), "v"(voff), "s"(src)
                : "memory");
            voff += (uint32_t)(THREADS * 16);
            ldsa += (uint32_t)(THREADS * 16);
        }
    }
    // Drain this wave's async copies, then make all waves' copies visible.
    asm volatile("s_wait_asynccnt 0x0" ::: "memory");
    __syncthreads();

    // ---- Pass 1: X_sum[l] = sum_d tile[l][d]  (deterministic reduction) ----
    #pragma unroll
    for (int l = 0; l < LAYERS; ++l) {
        const float* row = &tile[l * DIM];
        float a = row[tid] + row[tid + 256] + row[tid + 512];
        #pragma unroll
        for (int off = 16; off > 0; off >>= 1)
            a += __shfl_xor(a, off, 32);
        if (lane == 0) wpart[wave][l] = a;
    }
    __syncthreads();

    if (tid < LAYERS) {
        float s = 0.0f;
        #pragma unroll
        for (int w = 0; w < WAVES; ++w) s += wpart[w][tid];
        xsum[tid] = s;
    }
    __syncthreads();

#ifdef USE_WMMA_MLP
    // ---- MLP via V_WMMA_F32_16X16X4_F32, D = A*B + C ----
    // A = W (16x16, rows>=12 zero), per documented 16x4-f32 layout chunked over K:
    //     lane%16 = M, K = chunk*4 + 2*(lane>=16) + vgpr.
    // B = x broadcast over all N columns: B[k][n] = x[k]  (value independent of n,
    //     so every column of D holds the same matvec result).
    // D(VGPR r): lanes 0-15 -> h[r], lanes 16-31 -> h[r+8].
    // All waves execute the WMMA with full EXEC (no wave branch: ISA requires
    // EXEC all-ones and EXEC==0 does not skip WMMA); only wave0 lanes 0/16 commit.
    {
        const int m  = lane & 15;
        const int kh = lane >> 4;

        // Layer 1: h1 = tanh(W1 @ xsum + b1)
        v8f acc = {};
        #pragma unroll
        for (int c = 0; c < 3; ++c) {
            const int k0 = c * 4 + kh * 2;
            v2f a, b;
            a.x = (m < LAYERS) ? W1[m * LAYERS + k0]     : 0.0f;
            a.y = (m < LAYERS) ? W1[m * LAYERS + k0 + 1] : 0.0f;
            b.x = xsum[k0];
            b.y = xsum[k0 + 1];
            acc = __builtin_amdgcn_wmma_f32_16x16x4_f32(
                false, a, false, b, (short)0, acc, false, false);
        }
        if (tid == 0) {
            #pragma unroll
            for (int r = 0; r < 8; ++r) h1s[r] = tanhf(acc[r] + b1[r]);
        } else if (tid == 16) {
            #pragma unroll
            for (int r = 0; r < 4; ++r) h1s[r + 8] = tanhf(acc[r] + b1[r + 8]);
        }
    }
    __syncthreads();
    {
        const int m  = lane & 15;
        const int kh = lane >> 4;

        // Layer 2: h2 = tanh(W2 @ h1 + b2)  -> logits into wgt[]
        v8f acc = {};
        #pragma unroll
        for (int c = 0; c < 3; ++c) {
            const int k0 = c * 4 + kh * 2;
            v2f a, b;
            a.x = (m < LAYERS) ? W2[m * LAYERS + k0]     : 0.0f;
            a.y = (m < LAYERS) ? W2[m * LAYERS + k0 + 1] : 0.0f;
            b.x = h1s[k0];
            b.y = h1s[k0 + 1];
            acc = __builtin_amdgcn_wmma_f32_16x16x4_f32(
                false, a, false, b, (short)0, acc, false, false);
        }
        if (tid == 0) {
            #pragma unroll
            for (int r = 0; r < 8; ++r) wgt[r] = tanhf(acc[r] + b2[r]);
        } else if (tid == 16) {
            #pragma unroll
            for (int r = 0; r < 4; ++r) wgt[r + 8] = tanhf(acc[r] + b2[r + 8]);
        }
    }
    __syncthreads();
#else
    // ---- Scalar fallback MLP ----
    if (tid < LAYERS) {
        float acc = b1[tid];
        #pragma unroll
        for (int j = 0; j < LAYERS; ++j)
            acc = fmaf(W1[tid * LAYERS + j], xsum[j], acc);
        h1s[tid] = tanhf(acc);
    }
    __syncthreads();
    if (tid < LAYERS) {
        float acc = b2[tid];
        #pragma unroll
        for (int j = 0; j < LAYERS; ++j)
            acc = fmaf(W2[tid * LAYERS + j], h1s[j], acc);
        wgt[tid] = tanhf(acc);
    }
    __syncthreads();
#endif

    // ---- Softmax over 12 layers (serial on thread 0: trivial size) ----
    if (tid == 0) {
        float mx = wgt[0];
        #pragma unroll
        for (int l = 1; l < LAYERS; ++l) mx = fmaxf(mx, wgt[l]);
        float e[LAYERS];
        float s = 0.0f;
        #pragma unroll
        for (int l = 0; l < LAYERS; ++l) { e[l] = expf(wgt[l] - mx); s += e[l]; }
        float inv = 1.0f / s;
        #pragma unroll
        for (int l = 0; l < LAYERS; ++l) wgt[l] = e[l] * inv;
    }
    __syncthreads();

    // ---- Pass 2: out[d] = sum_l wgt[l] * tile[l][d] ----
    float w[LAYERS];
    #pragma unroll
    for (int l = 0; l < LAYERS; ++l) w[l] = wgt[l];

    float* __restrict__ dst = out + token * (size_t)DIM;
    #pragma unroll
    for (int r = 0; r < 3; ++r) {
        const int d = tid + r * 256;
        float acc = 0.0f;
        #pragma unroll
        for (int l = 0; l < LAYERS; ++l)
            acc = fmaf(w[l], tile[l * DIM + d], acc);
        dst[d] = acc;
    }
}

extern "C" void kernel_launch(void* const* d_in, const int* in_sizes, int n_in,
                              void* d_out, int out_size, void* d_ws, size_t ws_size,
                              hipStream_t stream) {
    const float* X  = (const float*)d_in[0];
    const float* W1 = (const float*)d_in[1];
    const float* b1 = (const float*)d_in[2];
    const float* W2 = (const float*)d_in[3];
    const float* b2 = (const float*)d_in[4];
    float* out = (float*)d_out;

    const int tokens = in_sizes[0] / TOKEN_ELEMS;   // B*S = 16384
    hipLaunchKernelGGL(msattn_kernel, dim3(tokens), dim3(THREADS), 0, stream,
                       X, W1, b1, W2, b2, out);
}